// TwoLayerGAT_16965120819895
// MI455X (gfx1250) — compile-verified
//
#include <hip/hip_runtime.h>

// ---------------------------------------------------------------------------
// Two-layer GAT for MI455X (gfx1250, wave32).
//  - GEMMs via V_WMMA_F32_16X16X4_F32 (full fp32: GEMMs are ~10 GFLOP total,
//    far from the bottleneck, so no precision downconversion is needed).
//  - Edge softmax via monotonic-uint atomicMax + f32 atomicAdd (L2-resident:
//    all scatter/gather targets fit in the 192 MB L2).
//  - Message passing: one wave per edge, alphas broadcast with wave32 shuffles.
// ---------------------------------------------------------------------------

typedef __attribute__((ext_vector_type(2))) float v2f;
typedef __attribute__((ext_vector_type(8))) float v8f;

#define N_NODES   100000
#define N_EDGES   1600000
#define EPTOT     (N_EDGES + N_NODES)   // edges + self loops
#define HEADS1    4
#define HID1      64
#define F_IN      128
#define F_MID     256                   // HEADS1 * HID1
#define F_OUT     64
#define NEG_SLOPE 0.2f

static_assert(N_NODES % 16 == 0, "row tiles must cover N exactly (EXEC all-1 for WMMA)");

// ---- helpers ---------------------------------------------------------------

// order-preserving float <-> uint mapping for atomicMax-based segment max
__device__ __forceinline__ unsigned fenc(float f) {
  unsigned u = __float_as_uint(f);
  return (u & 0x80000000u) ? ~u : (u | 0x80000000u);
}
__device__ __forceinline__ float fdec(unsigned k) {
  unsigned u = (k & 0x80000000u) ? (k & 0x7fffffffu) : ~k;
  return __uint_as_float(u);
}
__device__ __forceinline__ float lrelu(float x) { return x > 0.f ? x : NEG_SLOPE * x; }

// edge e in [0, EPTOT): first N_EDGES come from edge_index, rest are self loops
__device__ __forceinline__ void edge_ends(const int* __restrict__ ei, int e, int& s, int& d) {
  if (e < N_EDGES) { s = ei[e]; d = ei[N_EDGES + e]; }
  else             { s = e - N_EDGES; d = s; }
}

// ---- fp32 WMMA GEMM: OUT[N x ncols] = X[N x K] @ W[K x ncols] --------------
// One wave computes one 16x16 output tile; A tile staged in LDS (padded to
// avoid 16-way bank conflicts on the column reads).
template <int K>
__global__ void gemm_wmma_f32(const float* __restrict__ X,
                              const float* __restrict__ W,
                              float* __restrict__ OUT, int ncols) {
  constexpr int LDA = K + 2;
  __shared__ float As[16 * LDA];
  const int tid  = threadIdx.x;
  const int wave = tid >> 5;
  const int lane = tid & 31;
  const int row0 = blockIdx.x * 16;
  const int colTile = wave + (blockDim.x >> 5) * blockIdx.y;
  const int col0 = colTile * 16;

  // cooperative, coalesced load of the shared 16 x K A-tile
  for (int idx = tid; idx < 16 * K; idx += blockDim.x) {
    int r = idx / K, c = idx % K;
    As[r * LDA + c] = X[(size_t)(row0 + r) * K + c];
  }
  __syncthreads();

  // ISA A layout (16x4 f32): lanes 0-15 hold K=0,1 ; lanes 16-31 hold K=2,3
  const int m  = lane & 15;
  const int kh = (lane >> 4) << 1;   // 0 or 2
  v8f acc = {};
#pragma unroll 4
  for (int k = 0; k < K; k += 4) {
    v2f a, b;
    a.x = As[m * LDA + k + kh];
    a.y = As[m * LDA + k + kh + 1];
    b.x = W[(size_t)(k + kh) * ncols + col0 + m];
    b.y = W[(size_t)(k + kh + 1) * ncols + col0 + m];
    acc = __builtin_amdgcn_wmma_f32_16x16x4_f32(
        /*neg_a=*/false, a, /*neg_b=*/false, b,
        /*c_mod=*/(short)0, acc, /*reuse_a=*/false, /*reuse_b=*/false);
  }

  // C/D layout: VGPR r, lane l -> row (l>>4)*8 + r, col l&15
  const int orow = row0 + (lane >> 4) * 8;
  const int ocol = col0 + m;
#pragma unroll
  for (int r = 0; r < 8; ++r)
    OUT[(size_t)(orow + r) * ncols + ocol] = acc[r];
}

// ---- attention logits: out[n,h] = sum_c H[n, h*CHAN + c] * att[h, c] -------
template <int HEADS, int CHAN>
__global__ void attn_logits(const float* __restrict__ H,
                            const float* __restrict__ att,
                            float* __restrict__ out) {
  int i = blockIdx.x * blockDim.x + threadIdx.x;
  if (i >= N_NODES * HEADS) return;
  int node = i / HEADS, h = i - node * HEADS;
  const float* hp = H + (size_t)i * CHAN;          // (node*HEADS + h) * CHAN
  const float* ap = att + h * CHAN;
  float s = 0.f;
#pragma unroll 8
  for (int c = 0; c < CHAN; ++c) s += hp[c] * ap[c];
  out[i] = s;
}

// ---- edge softmax: segment max / segment sum over incoming edges -----------
template <int HEADS>
__global__ void edge_softmax_max(const int* __restrict__ ei,
                                 const float* __restrict__ a_s,
                                 const float* __restrict__ a_d,
                                 unsigned* __restrict__ menc) {
  int i = blockIdx.x * blockDim.x + threadIdx.x;
  if (i >= EPTOT * HEADS) return;
  int e = i / HEADS, h = i - e * HEADS;
  int s, d; edge_ends(ei, e, s, d);
  float v = lrelu(a_s[s * HEADS + h] + a_d[d * HEADS + h]);
  atomicMax(&menc[d * HEADS + h], fenc(v));
}

template <int HEADS>
__global__ void edge_softmax_sum(const int* __restrict__ ei,
                                 const float* __restrict__ a_s,
                                 const float* __restrict__ a_d,
                                 const unsigned* __restrict__ menc,
                                 float* __restrict__ z) {
  int i = blockIdx.x * blockDim.x + threadIdx.x;
  if (i >= EPTOT * HEADS) return;
  int e = i / HEADS, h = i - e * HEADS;
  int s, d; edge_ends(ei, e, s, d);
  float v  = lrelu(a_s[s * HEADS + h] + a_d[d * HEADS + h]);
  float mx = fdec(menc[d * HEADS + h]);
  atomicAdd(&z[d * HEADS + h], __expf(v - mx));
}

// ---- layer-1 message pass: OUT[dst, h*64+c] += H[src, h*64+c] * alpha[h] ---
// one wave per edge; lanes 0..3 compute the 4 head alphas, shfl-broadcast
__global__ void edge_message4(const int* __restrict__ ei,
                              const float* __restrict__ H,
                              const float* __restrict__ a_s,
                              const float* __restrict__ a_d,
                              const unsigned* __restrict__ menc,
                              const float* __restrict__ z,
                              float* __restrict__ OUT) {
  int wid  = (int)(((size_t)blockIdx.x * blockDim.x + threadIdx.x) >> 5);
  int lane = threadIdx.x & 31;
  if (wid >= EPTOT) return;
  int s, d; edge_ends(ei, wid, s, d);

  float alpha = 0.f;
  if (lane < HEADS1) {
    float v = lrelu(a_s[s * HEADS1 + lane] + a_d[d * HEADS1 + lane]);
    alpha = __expf(v - fdec(menc[d * HEADS1 + lane])) / (z[d * HEADS1 + lane] + 1e-16f);
  }
  float al[HEADS1];
#pragma unroll
  for (int h = 0; h < HEADS1; ++h) al[h] = __shfl(alpha, h, 32);

  const float* hs = H + (size_t)s * F_MID;
  float* od = OUT + (size_t)d * F_MID;
#pragma unroll
  for (int j = 0; j < 8; ++j) {
    int c = lane + j * 32;                 // head = c>>6 (constant per j)
    atomicAdd(&od[c], hs[c] * al[c >> 6]);
  }
}

// ---- layer-2 message pass (1 head, 64 channels): one wave per edge ---------
__global__ void edge_message1(const int* __restrict__ ei,
                              const float* __restrict__ H,
                              const float* __restrict__ a_s,
                              const float* __restrict__ a_d,
                              const unsigned* __restrict__ menc,
                              const float* __restrict__ z,
                              float* __restrict__ OUT) {
  int wid  = (int)(((size_t)blockIdx.x * blockDim.x + threadIdx.x) >> 5);
  int lane = threadIdx.x & 31;
  if (wid >= EPTOT) return;
  int s, d; edge_ends(ei, wid, s, d);
  float v = lrelu(a_s[s] + a_d[d]);
  float alpha = __expf(v - fdec(menc[d])) / (z[d] + 1e-16f);
  const float* hs = H + (size_t)s * F_OUT;
  float* od = OUT + (size_t)d * F_OUT;
  atomicAdd(&od[lane],      hs[lane]      * alpha);
  atomicAdd(&od[lane + 32], hs[lane + 32] * alpha);
}

// ---- elementwise kernels ---------------------------------------------------
__global__ void fill_u32(unsigned* __restrict__ p, unsigned v, size_t n) {
  size_t i = (size_t)blockIdx.x * blockDim.x + threadIdx.x;
  if (i < n) p[i] = v;
}
__global__ void init_bias_rows(float* __restrict__ out, const float* __restrict__ b,
                               size_t n, int width) {
  size_t i = (size_t)blockIdx.x * blockDim.x + threadIdx.x;
  if (i < n) out[i] = b[i % width];
}
__global__ void bias_elu(float* __restrict__ g, const float* __restrict__ b,
                         size_t n, int width) {
  size_t i = (size_t)blockIdx.x * blockDim.x + threadIdx.x;
  if (i >= n) return;
  float v = g[i] + b[i % width];
  g[i] = v > 0.f ? v : __expf(v) - 1.f;
}

// ---------------------------------------------------------------------------
extern "C" void kernel_launch(void* const* d_in, const int* in_sizes, int n_in,
                              void* d_out, int out_size, void* d_ws, size_t ws_size,
                              hipStream_t stream) {
  (void)in_sizes; (void)n_in; (void)out_size; (void)ws_size;

  const float* x    = (const float*)d_in[0];
  const int*   ei   = (const int*)  d_in[1];   // [2, E] flat
  const float* W1   = (const float*)d_in[2];
  const float* asr1 = (const float*)d_in[3];
  const float* ads1 = (const float*)d_in[4];
  const float* b1   = (const float*)d_in[5];
  const float* W2   = (const float*)d_in[6];
  const float* asr2 = (const float*)d_in[7];
  const float* ads2 = (const float*)d_in[8];
  const float* b2   = (const float*)d_in[9];
  float* out = (float*)d_out;

  // workspace layout (~238 MB)
  float* p = (float*)d_ws;
  float* h1 = p;           p += (size_t)N_NODES * F_MID;    // x @ W1
  float* g1 = p;           p += (size_t)N_NODES * F_MID;    // layer-1 aggregation / ELU
  float* hg = p;           p += (size_t)N_NODES * F_OUT;    // elu(g1) @ W2
  float* as1 = p;          p += (size_t)N_NODES * HEADS1;
  float* ad1 = p;          p += (size_t)N_NODES * HEADS1;
  float* z1  = p;          p += (size_t)N_NODES * HEADS1;
  unsigned* m1 = (unsigned*)p; p += (size_t)N_NODES * HEADS1;
  float* as2 = p;          p += (size_t)N_NODES;
  float* ad2 = p;          p += (size_t)N_NODES;
  float* z2  = p;          p += (size_t)N_NODES;
  unsigned* m2 = (unsigned*)p; p += (size_t)N_NODES;

  auto nb = [](size_t n, int b) { return (unsigned)((n + (size_t)b - 1) / b); };

  // deterministic re-init of all accumulators (fenc encoding: 0u == -inf side)
  fill_u32<<<nb((size_t)N_NODES * F_MID, 256), 256, 0, stream>>>((unsigned*)g1, 0u, (size_t)N_NODES * F_MID);
  fill_u32<<<nb((size_t)N_NODES * HEADS1, 256), 256, 0, stream>>>((unsigned*)z1, 0u, (size_t)N_NODES * HEADS1);
  fill_u32<<<nb((size_t)N_NODES * HEADS1, 256), 256, 0, stream>>>(m1, 0u, (size_t)N_NODES * HEADS1);
  fill_u32<<<nb((size_t)N_NODES, 256), 256, 0, stream>>>((unsigned*)z2, 0u, (size_t)N_NODES);
  fill_u32<<<nb((size_t)N_NODES, 256), 256, 0, stream>>>(m2, 0u, (size_t)N_NODES);
  init_bias_rows<<<nb((size_t)N_NODES * F_OUT, 256), 256, 0, stream>>>(out, b2, (size_t)N_NODES * F_OUT, F_OUT);

  // ---- layer 1 ----
  // 8 waves/block, each wave one 16x16 tile; grid.y covers 256 output cols
  gemm_wmma_f32<F_IN><<<dim3(N_NODES / 16, 2), 256, 0, stream>>>(x, W1, h1, F_MID);
  attn_logits<HEADS1, HID1><<<nb((size_t)N_NODES * HEADS1, 256), 256, 0, stream>>>(h1, asr1, as1);
  attn_logits<HEADS1, HID1><<<nb((size_t)N_NODES * HEADS1, 256), 256, 0, stream>>>(h1, ads1, ad1);
  edge_softmax_max<HEADS1><<<nb((size_t)EPTOT * HEADS1, 256), 256, 0, stream>>>(ei, as1, ad1, m1);
  edge_softmax_sum<HEADS1><<<nb((size_t)EPTOT * HEADS1, 256), 256, 0, stream>>>(ei, as1, ad1, m1, z1);
  edge_message4<<<nb((size_t)EPTOT * 32, 256), 256, 0, stream>>>(ei, h1, as1, ad1, m1, z1, g1);
  bias_elu<<<nb((size_t)N_NODES * F_MID, 256), 256, 0, stream>>>(g1, b1, (size_t)N_NODES * F_MID, F_MID);

  // ---- layer 2 ----
  // 4 waves/block cover the 64 output cols
  gemm_wmma_f32<F_MID><<<dim3(N_NODES / 16, 1), 128, 0, stream>>>(g1, W2, hg, F_OUT);
  attn_logits<1, F_OUT><<<nb((size_t)N_NODES, 256), 256, 0, stream>>>(hg, asr2, as2);
  attn_logits<1, F_OUT><<<nb((size_t)N_NODES, 256), 256, 0, stream>>>(hg, ads2, ad2);
  edge_softmax_max<1><<<nb((size_t)EPTOT, 256), 256, 0, stream>>>(ei, as2, ad2, m2);
  edge_softmax_sum<1><<<nb((size_t)EPTOT, 256), 256, 0, stream>>>(ei, as2, ad2, m2, z2);
  edge_message1<<<nb((size_t)EPTOT * 32, 256), 256, 0, stream>>>(ei, hg, as2, ad2, m2, z2, out);
}